// FT_NOLayer_11141145166393
// MI455X (gfx1250) — compile-verified
//
#include <hip/hip_runtime.h>
#include <math.h>

// v2f = A/B operand of V_WMMA_F32_16X16X4_F32 (16x4 f32 = 64 vals / 32 lanes = 2 VGPRs)
typedef __attribute__((ext_vector_type(2))) float v2f;
typedef __attribute__((ext_vector_type(8))) float v8f;

#define TWO_PI 6.283185307179586f

// One wave32 per point n.
//   fa   = lane&15            (M row of A and D: fa = f*4 + a)
//   koff = (lane<16) ? 0 : 2  (K half within each K=4 chunk, 32-bit A/B layout)
// 7 chunks of K=4 cover S=28 exactly; two f32 accumulators (re/im) share B = x.
__global__ __launch_bounds__(256)
void ftno_wmma_kernel(const float* __restrict__ x,            // (n, 28, 8)
                      const float* __restrict__ dists,        // (n, 28)
                      const float* __restrict__ angl,         // (n, 28)
                      const unsigned char* __restrict__ mask, // (n, 28) bool
                      const float* __restrict__ freqs,        // (4,)
                      const float* __restrict__ angles,       // (4,)
                      float* __restrict__ out,                // (n, 16, 8, 2)
                      float* __restrict__ mask_out,           // (n,)
                      int n_total)
{
    const int lane = threadIdx.x & 31;
    const int wave = threadIdx.x >> 5;
    const int n = blockIdx.x * (blockDim.x >> 5) + wave;
    if (n >= n_total) return;        // uniform per wave -> EXEC stays all-ones for WMMA

    const int fa   = lane & 15;      // row of A/D: fa = f*A + a
    const int f    = fa >> 2;
    const int a    = fa & 3;
    const int koff = (lane < 16) ? 0 : 2;

    const float fr   = fminf(freqs[f], 100.0f);      // MAX_FREQ clamp
    const float angp = angles[a];

    const float*         dN = dists + n * 28;
    const float*         aN = angl  + n * 28;
    const unsigned char* mN = mask  + n * 28;
    const float*         xN = x     + (size_t)n * 224;

    // ---- Preload everything this lane needs (straight-line, no exec branching).
    // B column: lanes with fa>=8 are zero-padding columns; they re-read column
    // fa&7 (L1 broadcast) and zero it with one multiply instead of a branch.
    const int   c7    = fa & 7;
    const float cmask = (fa < 8) ? 1.0f : 0.0f;

    float dv[14], av[14], xv[14];
    int   mv[14];
#pragma unroll
    for (int i = 0; i < 14; ++i) {
        const int s = (i >> 1) * 4 + koff + (i & 1);   // K index = s in [0,28)
        dv[i] = dN[s];
        av[i] = aN[s];
        mv[i] = (int)mN[s];                            // u8 load, zero-extended
        xv[i] = xN[s * 8 + c7];
    }
#pragma unroll
    for (int i = 0; i < 14; ++i) xv[i] *= cmask;       // zero padding columns

    // ---- Phase 1: this lane's 14 complex weights + abs-sum (pure VALU) ----
    float wre[14], wim[14];
    float abssum = 0.0f;
#pragma unroll
    for (int i = 0; i < 14; ++i) {
        const float dir = __cosf(av[i] - angp);        // cos(angl - theta_a)
        const float p   = dv[i] * fr * dir;
        float sre, sim;
        __sincosf(TWO_PI * p, &sim, &sre);             // exp(i*2*pi*p)
        const float wm = 1.0f - (float)mv[i];          // mask arithmetically
        sre = fmaf(sre, wm, 1e-20f);                   // w*(~m) + epsilon
        sim = sim * wm;
        wre[i] = sre;
        wim[i] = sim;
        abssum += sqrtf(sre * sre + sim * sim);        // |w| for normalization
    }
    // lanes (l, l^16) hold the two K-halves of the same fa row -> full denominator
    abssum += __shfl_xor(abssum, 16, 32);

    // fast reciprocal: v_rcp_f32 + one Newton step (~0.5 ulp)
    float inv = __builtin_amdgcn_rcpf(abssum);
    inv = inv * (2.0f - abssum * inv);
#pragma unroll
    for (int i = 0; i < 14; ++i) { wre[i] *= inv; wim[i] *= inv; }

    // ---- Phase 2: D += A x B on the f32 matrix pipe, K=4 per step, 7 steps ----
    v8f acc_re = {};
    v8f acc_im = {};
#pragma unroll
    for (int chunk = 0; chunk < 7; ++chunk) {
        v2f are, aim, bb;
        are.x = wre[2 * chunk];     are.y = wre[2 * chunk + 1];
        aim.x = wim[2 * chunk];     aim.y = wim[2 * chunk + 1];
        bb.x  = xv[2 * chunk];      bb.y  = xv[2 * chunk + 1];
        // v_wmma_f32_16x16x4_f32: (neg_a, A, neg_b, B, c_mod, C, reuse_a, reuse_b)
        acc_re = __builtin_amdgcn_wmma_f32_16x16x4_f32(
            false, are, false, bb, (short)0, acc_re, false, false);
        acc_im = __builtin_amdgcn_wmma_f32_16x16x4_f32(
            false, aim, false, bb, (short)0, acc_im, false, false);
    }

    // ---- Store: D VGPR r holds row fa = r + (lane<16?0:8), column = lane&15 ----
    if (fa < 8) {
        const int mbase = (lane < 16) ? 0 : 8;
        float* o = out + (size_t)n * 256;              // (16 fa) * (8 c) * (2 re/im)
#pragma unroll
        for (int r = 0; r < 8; ++r) {
            const int fao = mbase + r;
            float2 v;
            v.x = acc_re[r];
            v.y = acc_im[r];
            *(float2*)(o + (fao * 8 + fa) * 2) = v;    // interleaved {re, im}
        }
    }

    // ---- mask_out: all 28 masked <=> abssum == 28e-20 << 0.5; any unmasked
    //      neighbor contributes |exp(i*theta)| ~= 1 >= 0.5. One compare. ----
    if (lane == 0) {
        mask_out[n] = (abssum < 0.5f) ? 1.0f : 0.0f;
    }
}

extern "C" void kernel_launch(void* const* d_in, const int* in_sizes, int n_in,
                              void* d_out, int out_size, void* d_ws, size_t ws_size,
                              hipStream_t stream) {
    (void)n_in; (void)out_size; (void)d_ws; (void)ws_size;
    const float*         x      = (const float*)d_in[0];
    const float*         dists  = (const float*)d_in[1];
    const float*         angl   = (const float*)d_in[2];
    const unsigned char* mask   = (const unsigned char*)d_in[3];  // jax bool = 1 byte
    const float*         freqs  = (const float*)d_in[4];
    const float*         angles = (const float*)d_in[5];

    const int n = in_sizes[1] / 28;                    // dists is (n, 4, 7)
    float* out  = (float*)d_out;                       // n*256 floats
    float* mout = out + (size_t)n * 256;               // then n mask_out floats

    dim3 block(256);                                   // 8 waves = 8 points per block
    dim3 grid((unsigned)((n + 7) / 8));
    hipLaunchKernelGGL(ftno_wmma_kernel, grid, block, 0, stream,
                       x, dists, angl, mask, freqs, angles, out, mout, n);
}